// Trojan_GCN_38895223833221
// MI455X (gfx1250) — compile-verified
//
#include <hip/hip_runtime.h>

typedef __attribute__((ext_vector_type(2))) float v2f;
typedef __attribute__((ext_vector_type(8))) float v8f;

// ---------------------------------------------------------------------------
// zero-fill (grid-stride)
// ---------------------------------------------------------------------------
__global__ void gcn_zero_kernel(float* __restrict__ p, long n) {
    long i = (long)blockIdx.x * blockDim.x + threadIdx.x;
    long stride = (long)gridDim.x * blockDim.x;
    for (; i < n; i += stride) p[i] = 0.0f;
}

// ---------------------------------------------------------------------------
// degree accumulation: deg_out over src, deg_in over dst
// ---------------------------------------------------------------------------
__global__ void gcn_degree_kernel(const int* __restrict__ src,
                                  const int* __restrict__ dst,
                                  float* __restrict__ degOut,
                                  float* __restrict__ degIn, int E) {
    int i = blockIdx.x * blockDim.x + threadIdx.x;
    if (i < E) {
        unsafeAtomicAdd(&degOut[src[i]], 1.0f);
        unsafeAtomicAdd(&degIn[dst[i]], 1.0f);
    }
}

// in place: deg -> clip(deg,1)^-0.5
__global__ void gcn_norm_kernel(float* __restrict__ p, int n) {
    int i = blockIdx.x * blockDim.x + threadIdx.x;
    if (i < n) {
        float d = p[i];
        d = d < 1.0f ? 1.0f : d;
        p[i] = 1.0f / sqrtf(d);
    }
}

// ---------------------------------------------------------------------------
// edge scatter: agg[dst] += norm_out[src] * x[src]   (float4 per thread)
// shift = log2(d/4), mask = d/4 - 1
// ---------------------------------------------------------------------------
__global__ void gcn_scatter_kernel(const float* __restrict__ x,
                                   const int* __restrict__ src,
                                   const int* __restrict__ dst,
                                   const float* __restrict__ normOut,
                                   float* __restrict__ agg,
                                   long total, int shift, int mask, int dchan) {
    long gid = (long)blockIdx.x * blockDim.x + threadIdx.x;
    if (gid >= total) return;
    int e = (int)(gid >> shift);
    int c = ((int)gid & mask) << 2;
    int s = src[e];
    int t = dst[e];
    float sc = normOut[s];
    const float4 v = *(const float4*)(x + (long)s * dchan + c);
    float* ap = agg + (long)t * dchan + c;
    unsafeAtomicAdd(ap + 0, v.x * sc);
    unsafeAtomicAdd(ap + 1, v.y * sc);
    unsafeAtomicAdd(ap + 2, v.z * sc);
    unsafeAtomicAdd(ap + 3, v.w * sc);
}

// ---------------------------------------------------------------------------
// WMMA fp32 GEMM:  C = [relu]( scale_rows(A_gathered) @ W + bias )
//   A: [M,K] row-major (row m taken from rowIndex[m] if given)
//   W: [K,Nout] row-major,  bias: [Nout]
//   rowScale: optional per-row multiplier on A (norm_in folding)
// One wave (32 lanes) per 16x32 output tile: two accumulators share each
// A fragment, so every A load feeds TWO v_wmma_f32_16x16x4_f32 ops.
// Requires: M mult of 16, Nout mult of 32, K mult of 4 (true for all layers).
// ---------------------------------------------------------------------------
__global__ void gcn_gemm_wmma_kernel(const float* __restrict__ A,
                                     const float* __restrict__ W,
                                     const float* __restrict__ bias,
                                     float* __restrict__ C,
                                     int M, int K, int Nout,
                                     const float* __restrict__ rowScale,
                                     const int* __restrict__ rowIndex,
                                     int applyRelu) {
    const int lane = threadIdx.x;       // 0..31
    const int half = lane >> 4;         // 0 or 1
    const int l    = lane & 15;

    const int tileM = blockIdx.y * blockDim.y + threadIdx.y;  // wave-uniform
    const int tileN = blockIdx.x;                             // 32-wide tiles
    if (tileM * 16 >= M) return;        // uniform per wave -> EXEC stays all-1
    const int m0 = tileM * 16;
    const int n0 = tileN * 32;

    // A-operand layout (16x4 f32): lanes 0-15 hold rows m0+l at K=kb+0,kb+1;
    // lanes 16-31 hold the same rows at K=kb+2,kb+3.
    const int arow = m0 + l;
    const int srcRow = rowIndex ? rowIndex[arow] : arow;
    const float* __restrict__ Arow = A + (long)srcRow * K;
    const float sc = rowScale ? rowScale[arow] : 1.0f;

    const float* __restrict__ Bcol0 = W + n0 + l;        // cols n0   .. n0+15
    const float* __restrict__ Bcol1 = W + n0 + 16 + l;   // cols n0+16.. n0+31

    v8f acc0 = {};
    v8f acc1 = {};
    for (int kb = 0; kb < K; kb += 4) {
        const int ka = kb + 2 * half;
        v2f a, b0, b1;
        a.x  = Arow[ka]     * sc;
        a.y  = Arow[ka + 1] * sc;
        b0.x = Bcol0[(long)ka * Nout];
        b0.y = Bcol0[(long)(ka + 1) * Nout];
        b1.x = Bcol1[(long)ka * Nout];
        b1.y = Bcol1[(long)(ka + 1) * Nout];
        acc0 = __builtin_amdgcn_wmma_f32_16x16x4_f32(
                   false, a, false, b0, (short)0, acc0, false, false);
        acc1 = __builtin_amdgcn_wmma_f32_16x16x4_f32(
                   false, a, false, b1, (short)0, acc1, false, false);
    }

    const float bv0 = bias[n0 + l];
    const float bv1 = bias[n0 + 16 + l];
    float* __restrict__ Cout0 = C + n0 + l;
    float* __restrict__ Cout1 = C + n0 + 16 + l;
#pragma unroll
    for (int i = 0; i < 8; ++i) {
        const long row = (long)(m0 + i + 8 * half) * Nout;
        float v0 = acc0[i] + bv0;
        float v1 = acc1[i] + bv1;
        if (applyRelu) {
            v0 = v0 > 0.0f ? v0 : 0.0f;
            v1 = v1 > 0.0f ? v1 : 0.0f;
        }
        Cout0[row] = v0;
        Cout1[row] = v1;
    }
}

// ---------------------------------------------------------------------------
// launch
// ---------------------------------------------------------------------------
extern "C" void kernel_launch(void* const* d_in, const int* in_sizes, int n_in,
                              void* d_out, int out_size, void* d_ws, size_t ws_size,
                              hipStream_t stream) {
    const float* features = (const float*)d_in[0];
    const int*   src      = (const int*)d_in[1];
    const int*   dst      = (const int*)d_in[2];
    const int*   trig     = (const int*)d_in[3];
    const float* W1  = (const float*)d_in[4];   // [128,256]
    const float* b1  = (const float*)d_in[5];
    const float* W2  = (const float*)d_in[6];   // [256,128]
    const float* b2  = (const float*)d_in[7];
    const float* Wf  = (const float*)d_in[8];   // [128,128]
    const float* bf  = (const float*)d_in[9];
    const float* Wf1 = (const float*)d_in[10];  // [128,128]
    const float* bf1 = (const float*)d_in[11];
    const float* We  = (const float*)d_in[12];  // [128,64]
    const float* be  = (const float*)d_in[13];
    const float* We1 = (const float*)d_in[14];  // [64,64]
    const float* be1 = (const float*)d_in[15];

    const int d  = 128;
    const int h  = 256;
    const int od = 64;
    const int N  = in_sizes[0] / d;
    const int E  = in_sizes[1];
    const int T  = in_sizes[3];

    float* ws     = (float*)d_ws;
    float* degOut = ws;                          // [N] -> norm_out
    float* degIn  = ws + N;                      // [N] -> norm_in
    float* agg1   = ws + 2L * N;                 // [N,128], later reused as h2
    float* h1     = agg1 + (long)N * d;          // [N,256]
    float* agg2   = h1 + (long)N * h;            // [N,256]
    float* tmp    = agg2 + (long)N * h;          // [T,128] head scratch
    float* h2     = agg1;                        // reuse

    float* outFeat = (float*)d_out;              // [T,128]
    float* outEdge = outFeat + (long)T * d;      // [T,64]

    // 1. zero degrees + agg1 + h1 + agg2
    {
        long zn = 2L * N + (long)N * d + 2L * (long)N * h;
        int blocks = (int)((zn + 255) / 256);
        if (blocks > (1 << 20)) blocks = (1 << 20);
        gcn_zero_kernel<<<blocks, 256, 0, stream>>>(ws, zn);
    }
    // 2. degrees
    gcn_degree_kernel<<<(E + 255) / 256, 256, 0, stream>>>(src, dst, degOut, degIn, E);
    // 3. norms (in place over both arrays)
    gcn_norm_kernel<<<(2 * N + 255) / 256, 256, 0, stream>>>(ws, 2 * N);

    // 4. layer-1 scatter: agg1[dst] += norm_out[src] * features[src]
    {
        long total = (long)E * (d / 4);
        gcn_scatter_kernel<<<(int)((total + 255) / 256), 256, 0, stream>>>(
            features, src, dst, degOut, agg1, total, 5, 31, d);
    }
    // 5. h1 = relu( (norm_in * agg1) @ W1 + b1 )   [N,256]
    {
        dim3 blk(32, 4), grd(h / 32, (N / 16 + 3) / 4);
        gcn_gemm_wmma_kernel<<<grd, blk, 0, stream>>>(agg1, W1, b1, h1, N, d, h,
                                                      degIn, nullptr, 1);
    }
    // 6. layer-2 scatter: agg2[dst] += norm_out[src] * h1[src]
    {
        long total = (long)E * (h / 4);
        gcn_scatter_kernel<<<(int)((total + 255) / 256), 256, 0, stream>>>(
            h1, src, dst, degOut, agg2, total, 6, 63, h);
    }
    // 7. h2 = relu( (norm_in * agg2) @ W2 + b2 )   [N,128]  (into agg1 space)
    {
        dim3 blk(32, 4), grd(d / 32, (N / 16 + 3) / 4);
        gcn_gemm_wmma_kernel<<<grd, blk, 0, stream>>>(agg2, W2, b2, h2, N, h, d,
                                                      degIn, nullptr, 1);
    }
    // 8. t1 = relu( h2[trig] @ Wf + bf )           [T,128]
    {
        dim3 blk(32, 4), grd(d / 32, (T / 16 + 3) / 4);
        gcn_gemm_wmma_kernel<<<grd, blk, 0, stream>>>(h2, Wf, bf, tmp, T, d, d,
                                                      nullptr, trig, 1);
    }
    // 9. feat = t1 @ Wf1 + bf1                     [T,128] -> d_out
    {
        dim3 blk(32, 4), grd(d / 32, (T / 16 + 3) / 4);
        gcn_gemm_wmma_kernel<<<grd, blk, 0, stream>>>(tmp, Wf1, bf1, outFeat, T, d, d,
                                                      nullptr, nullptr, 0);
    }
    // 10. t2 = relu( h2[trig] @ We + be )          [T,64]  (reuse tmp)
    {
        dim3 blk(32, 4), grd(od / 32, (T / 16 + 3) / 4);
        gcn_gemm_wmma_kernel<<<grd, blk, 0, stream>>>(h2, We, be, tmp, T, d, od,
                                                      nullptr, trig, 1);
    }
    // 11. edge_weight = t2 @ We1 + be1             [T,64] -> d_out + T*128
    {
        dim3 blk(32, 4), grd(od / 32, (T / 16 + 3) / 4);
        gcn_gemm_wmma_kernel<<<grd, blk, 0, stream>>>(tmp, We1, be1, outEdge, T, od, od,
                                                      nullptr, nullptr, 0);
    }
}